// BehlerG2_26431228739598
// MI455X (gfx1250) — compile-verified
//
#include <hip/hip_runtime.h>
#include <math.h>

typedef __attribute__((ext_vector_type(2))) float v2f;
typedef __attribute__((ext_vector_type(8))) float v8f;

#define TT    512   // triples per (b,n)
#define EE    16    // etas (M dim of WMMA)
#define NWAVE 4     // waves per block
#define TILE  32    // triples staged per wave per iteration
#define ROWP  34    // padded LDS row stride in floats (even -> 8B-aligned pairs, spreads banks)

// Branchless Behler cosine cutoff: compute unconditionally, select with v_cndmask.
// (all lanes are in-range for this workload, so exec-masked branches only serialize
// the three v_cos_f32 evaluations and add SALU churn)
__device__ __forceinline__ float cutf(float r) {
    const float k = 3.14159265358979323846f / 6.0f;   // pi / RC
    float c = 0.5f * (__cosf(r * k) + 1.0f);
    return (r < 6.0f) ? c : 0.0f;
}

// pow with exact repeated-squaring fast path for small integral exponents
// (zetas are wave-uniform scalars, so the branch is uniform)
__device__ __forceinline__ float zpow(float x, float z) {
    int zi = (int)z;
    if ((float)zi == z && zi >= 0 && zi <= 64) {
        float r = 1.0f, p = x;
        int n = zi;
        while (n) { if (n & 1) r *= p; p *= p; n >>= 1; }
        return r;
    }
    return powf(x, z);
}

__global__ __launch_bounds__(128)
void behler_angular_wmma(const float* __restrict__ r_ij,
                         const float* __restrict__ r_ik,
                         const float* __restrict__ r_jk,
                         const float* __restrict__ mask,
                         const float* __restrict__ etas,
                         const float* __restrict__ zetas,
                         float* __restrict__ out) {
    // wave-private staging tiles: Rt = R^T (16 etas x 32 triples),
    // At = G^T (16 cols x 32 triples, rows 8..15 stay zero = N-padding)
    __shared__ float Rt[NWAVE][EE][ROWP];
    __shared__ float At[NWAVE][EE][ROWP];
    __shared__ float obuf[16][8];

    const int tid  = threadIdx.x;
    const int lane = tid & 31;
    const int wave = tid >> 5;
    const int m    = lane & 15;   // M row (A) / N col (B) owned by this lane
    const int h    = lane >> 4;   // half-wave -> K pair {0,1} vs {2,3}

    // uniform constants
    float eta[EE];
    #pragma unroll
    for (int e = 0; e < EE; ++e) eta[e] = etas[e];
    float zt[4], s0[4], s1[4];
    #pragma unroll
    for (int z = 0; z < 4; ++z) {
        zt[z] = zetas[z];
        s0[z] = exp2f(1.0f - zt[z]);   // 2^(1-zeta)
        s1[z] = exp2f(1.0f + zt[z]);   // 2^(1+zeta)
    }

    // zero the B-matrix padding rows (cols 8..15) and the output buffer
    for (int i = lane; i < 8 * ROWP; i += 32)
        (&At[wave][8][0])[i] = 0.0f;
    ((float*)obuf)[tid & 127] = 0.0f;
    __syncthreads();

    const size_t base = (size_t)blockIdx.x * TT;
    v8f acc = {0.f, 0.f, 0.f, 0.f, 0.f, 0.f, 0.f, 0.f};

    #pragma unroll
    for (int it = 0; it < TT / (NWAVE * TILE); ++it) {
        const int t = wave * (TT / NWAVE) + it * TILE + lane;

        const float rij = r_ij[base + t];
        const float rik = r_ik[base + t];
        const float rjk = r_jk[base + t];
        const float mk  = mask[base + t];

        const float r2  = rij * rij + rik * rik + rjk * rjk;
        const float cut = cutf(rij) * cutf(rik) * cutf(rjk);
        const float mc  = mk * cut;                               // mask folded into radius part
        const float ct  = (mk == 0.0f) ? 0.0f : (r2 * 0.5f / (rij * rik));
        const float bse = 1.0f - ct;

        // scatter one triple-column of R^T and G^T into this wave's tiles
        #pragma unroll
        for (int e = 0; e < EE; ++e)
            Rt[wave][e][lane] = mc * __expf(-r2 * eta[e]);
        #pragma unroll
        for (int z = 0; z < 4; ++z) {
            const float p = zpow(bse, zt[z]);
            At[wave][z    ][lane] = s0[z] * p;
            At[wave][z + 4][lane] = s1[z] * p;
        }

        // in-wave cross-lane visibility: wait for our ds_stores, fence compiler
        asm volatile("s_wait_dscnt 0" ::: "memory");

        // 8 chained f32 WMMAs consume the 32-triple tile (K = 8 * 4)
        #pragma unroll
        for (int c = 0; c < 8; ++c) {
            const float* pr = &Rt[wave][m][c * 4 + 2 * h];
            const float* pa = &At[wave][m][c * 4 + 2 * h];
            v2f a; a.x = pr[0]; a.y = pr[1];
            v2f b; b.x = pa[0]; b.y = pa[1];
            acc = __builtin_amdgcn_wmma_f32_16x16x4_f32(
                false, a, false, b, (short)0, acc, false, false);
        }
        // keep next iteration's tile stores after this iteration's tile loads
        asm volatile("" ::: "memory");
    }

    // cross-wave reduction of the 16x8 useful block of each accumulator.
    // C/D layout: VGPR v holds row M = v + 8*h, column n = lane&15.
    if (m < 8) {
        #pragma unroll
        for (int v = 0; v < 8; ++v)
            atomicAdd(&obuf[v + 8 * h][m], acc[v]);
    }
    __syncthreads();

    // obuf[e][z'] flattens exactly to the (E*2Z) output layout
    out[(size_t)blockIdx.x * 128 + tid] = ((const float*)obuf)[tid];
}

extern "C" void kernel_launch(void* const* d_in, const int* in_sizes, int n_in,
                              void* d_out, int out_size, void* d_ws, size_t ws_size,
                              hipStream_t stream) {
    const float* r_ij = (const float*)d_in[0];
    const float* r_ik = (const float*)d_in[1];
    const float* r_jk = (const float*)d_in[2];
    const float* mask = (const float*)d_in[3];
    const float* etas = (const float*)d_in[4];
    const float* zetas = (const float*)d_in[5];
    float* out = (float*)d_out;

    const int nBN = out_size / 128;   // B*N blocks, one (b,n) pair each
    behler_angular_wmma<<<nBN, 128, 0, stream>>>(r_ij, r_ik, r_jk, mask, etas, zetas, out);
}